// GatedSpikingElasticReservoirLayer_9852654977359
// MI455X (gfx1250) — compile-verified
//
#include <hip/hip_runtime.h>
#include <math.h>

#define N_RES     4096
#define IN_DIM    1024
#define MAX_DIM   4608
#define BATCH     128
#define T_STEPS   128
#define LEAK      0.1f
#define THRESH    0.5f

#define CK        128          // K-chunk staged in LDS
#define LDA       132          // padded LDS row stride (floats): 132%64=4 -> 16 rows hit distinct banks; 528B is 16B aligned
#define NT        2            // 16x16 N-tiles per wave
#define NSTRIP    (NT * 16)    // 32 output columns per block

typedef __attribute__((ext_vector_type(2))) float v2f;
typedef __attribute__((ext_vector_type(4))) float v4f;
typedef __attribute__((ext_vector_type(8))) float v8f;

__device__ __forceinline__ float sigmoid_f(float x) {
    return 1.0f / (1.0f + expf(-x));
}

#define WMMA_F32(A, B, C) \
    __builtin_amdgcn_wmma_f32_16x16x4_f32(false, (A), false, (B), (short)0, (C), false, false)

// Fused per-timestep cell. Grid: 128 blocks (one 32-col N strip) x 256 threads
// (8 waves cover all 8 M-tiles, M = B = 128). Each wave owns a 16x32 output tile.
__global__ __launch_bounds__(256)
void reservoir_step_kernel(const float* __restrict__ prev,    // (B, MAX_DIM), first N cols live
                           const float* __restrict__ x_t,     // (B, IN_DIM)
                           const float* __restrict__ W_res,   // (N, N) row-major
                           const float* __restrict__ W_in,    // (N, IN_DIM) row-major
                           const float* __restrict__ W_gate,  // (3N, IN_DIM) row-major
                           float* __restrict__ out_t)         // (B, MAX_DIM)
{
    __shared__ float As[BATCH * LDA];       // staged A chunk (prev rows, then x_t rows)
    __shared__ float Bs[4 * NSTRIP * LDA];  // staged transposed weight strips (loop 2)

    const int tid   = threadIdx.x;
    const int lane  = tid & 31;
    const int wave  = tid >> 5;
    const int half  = lane >> 4;   // 0: lanes 0-15, 1: lanes 16-31
    const int idx   = lane & 15;
    const int mBase = wave * 16;
    const int nBase = blockIdx.x * NSTRIP;

    v8f acc[NT] = {};  // input_part + reservoir_part
    v8f gi[NT]  = {};
    v8f gf[NT]  = {};
    v8f go[NT]  = {};

    // ---------------- Loop 1: acc += prev[:, :N] @ W_res[:, nstrip] ----------------
    // A fragment comes from LDS (coalesced staging); B (W_res) is already coalesced
    // from global: 16 consecutive columns per lane group, rows k -> 16KB stride, L2-resident.
    for (int kc = 0; kc < N_RES; kc += CK) {
        __syncthreads();
        #pragma unroll
        for (int j = 0; j < (BATCH * CK / 4) / 256; ++j) {   // 16 float4 per thread
            const int i4  = tid + j * 256;
            const int row = i4 >> 5;          // CK/4 = 32 float4 per row
            const int k4  = i4 & 31;
            v4f v = *(const v4f*)(prev + (size_t)row * MAX_DIM + kc + k4 * 4);
            *(v4f*)(&As[row * LDA + k4 * 4]) = v;
        }
        __syncthreads();
        #pragma unroll 4
        for (int k = 0; k < CK; k += 4) {
            const int ao = (mBase + idx) * LDA + k + half * 2;
            v2f a; a.x = As[ao]; a.y = As[ao + 1];
            #pragma unroll
            for (int j = 0; j < NT; ++j) {
                const float* bp = W_res + (size_t)(kc + k + half * 2) * N_RES
                                        + (nBase + j * 16 + idx);
                v2f b; b.x = bp[0]; b.y = bp[N_RES];
                acc[j] = WMMA_F32(a, b, acc[j]);
            }
        }
    }

    // ---------------- Loop 2: acc += x_t @ W_in^T ; gates = x_t @ W_gate^T ----------------
    // Both A (x_t) and B (transposed weight rows) staged in LDS; one x_t fragment
    // feeds 4 matrices x NT n-tiles = 8 WMMAs.
    const float* wsrc[4] = {
        W_in   + (size_t)nBase * IN_DIM,
        W_gate + (size_t)nBase * IN_DIM,
        W_gate + (size_t)(N_RES + nBase) * IN_DIM,
        W_gate + (size_t)(2 * N_RES + nBase) * IN_DIM
    };
    for (int kc = 0; kc < IN_DIM; kc += CK) {
        __syncthreads();
        #pragma unroll
        for (int j = 0; j < (BATCH * CK / 4) / 256; ++j) {   // stage x_t chunk
            const int i4  = tid + j * 256;
            const int row = i4 >> 5;
            const int k4  = i4 & 31;
            v4f v = *(const v4f*)(x_t + (size_t)row * IN_DIM + kc + k4 * 4);
            *(v4f*)(&As[row * LDA + k4 * 4]) = v;
        }
        #pragma unroll
        for (int j = 0; j < (4 * NSTRIP * CK / 4) / 256; ++j) {  // stage 4 weight strips
            const int i4 = tid + j * 256;
            const int r  = i4 >> 5;       // 0..127: m = r>>5, n = r&31
            const int k4 = i4 & 31;
            const int m  = r >> 5;
            const int n  = r & 31;
            v4f v = *(const v4f*)(wsrc[m] + (size_t)n * IN_DIM + kc + k4 * 4);
            *(v4f*)(&Bs[r * LDA + k4 * 4]) = v;
        }
        __syncthreads();
        #pragma unroll 2
        for (int k = 0; k < CK; k += 4) {
            const int ao = (mBase + idx) * LDA + k + half * 2;
            v2f a; a.x = As[ao]; a.y = As[ao + 1];
            #pragma unroll
            for (int j = 0; j < NT; ++j) {
                const int colr = j * 16 + idx;
                const int ko   = k + half * 2;
                v2f b0, b1, b2, b3;
                b0.x = Bs[(0 * NSTRIP + colr) * LDA + ko]; b0.y = Bs[(0 * NSTRIP + colr) * LDA + ko + 1];
                b1.x = Bs[(1 * NSTRIP + colr) * LDA + ko]; b1.y = Bs[(1 * NSTRIP + colr) * LDA + ko + 1];
                b2.x = Bs[(2 * NSTRIP + colr) * LDA + ko]; b2.y = Bs[(2 * NSTRIP + colr) * LDA + ko + 1];
                b3.x = Bs[(3 * NSTRIP + colr) * LDA + ko]; b3.y = Bs[(3 * NSTRIP + colr) * LDA + ko + 1];
                acc[j] = WMMA_F32(a, b0, acc[j]);
                gi[j]  = WMMA_F32(a, b1, gi[j]);
                gf[j]  = WMMA_F32(a, b2, gf[j]);
                go[j]  = WMMA_F32(a, b3, go[j]);
            }
        }
    }

    // ---------------- Epilogue: C/D layout -> VGPR v = row (mBase + v + half*8), col idx ----
    #pragma unroll
    for (int j = 0; j < NT; ++j) {
        const int col = nBase + j * 16 + idx;
        #pragma unroll
        for (int v = 0; v < 8; ++v) {
            const int row = mBase + v + half * 8;
            const float pv  = prev[(size_t)row * MAX_DIM + col];
            const float igs = sigmoid_f(gi[j][v]);
            const float fgs = sigmoid_f(gf[j][v]);
            const float ogs = sigmoid_f(go[j][v]);
            float s = (1.0f - LEAK) * (fgs * pv) + LEAK * tanhf(igs * acc[j][v]);
            s *= ogs;
            if (s > THRESH) s -= THRESH;
            out_t[(size_t)row * MAX_DIM + col] = s;
        }
    }
}

// Zero the padding columns [N_RES, MAX_DIM) for all (t, b) rows once.
__global__ void zero_pad_kernel(float* __restrict__ out)
{
    const size_t padw  = MAX_DIM - N_RES;  // 512
    const size_t total = (size_t)T_STEPS * BATCH * padw;
    size_t i = (size_t)blockIdx.x * blockDim.x + threadIdx.x;
    if (i >= total) return;
    const size_t c  = i % padw;
    const size_t rb = i / padw;            // flat (t*B + b) row
    out[rb * MAX_DIM + N_RES + c] = 0.0f;
}

extern "C" void kernel_launch(void* const* d_in, const int* in_sizes, int n_in,
                              void* d_out, int out_size, void* d_ws, size_t ws_size,
                              hipStream_t stream) {
    const float* x      = (const float*)d_in[0];   // (T, B, IN_DIM)
    const float* state0 = (const float*)d_in[1];   // (B, MAX_DIM) zeros
    const float* W_res  = (const float*)d_in[2];   // (N, N)
    const float* W_in   = (const float*)d_in[3];   // (N, IN_DIM)
    const float* W_gate = (const float*)d_in[4];   // (3N, IN_DIM)
    float* out = (float*)d_out;                    // (T, B, MAX_DIM)

    const size_t pad_total = (size_t)T_STEPS * BATCH * (MAX_DIM - N_RES);
    zero_pad_kernel<<<(unsigned)((pad_total + 255) / 256), 256, 0, stream>>>(out);

    const float* prev = state0;
    for (int t = 0; t < T_STEPS; ++t) {
        const float* x_t  = x   + (size_t)t * BATCH * IN_DIM;
        float*       outt = out + (size_t)t * BATCH * MAX_DIM;
        reservoir_step_kernel<<<dim3(N_RES / NSTRIP), dim3(256), 0, stream>>>(
            prev, x_t, W_res, W_in, W_gate, outt);
        prev = outt;
    }
}